// DiffusionGraphConv_87565793230931
// MI455X (gfx1250) — compile-verified
//
#include <hip/hip_runtime.h>

typedef float v2f __attribute__((ext_vector_type(2)));
typedef float v8f __attribute__((ext_vector_type(8)));

#define NN 50000
#define NE 800000
#define F  64
#define KD 256   // F * (NUM_HOPS + 1)

// ---------------- graph message-passing kernels ----------------

__global__ __launch_bounds__(256) void zero_k(float* __restrict__ p, long n) {
    long i  = (long)blockIdx.x * blockDim.x + threadIdx.x;
    long st = (long)gridDim.x * blockDim.x;
    for (; i < n; i += st) p[i] = 0.0f;
}

__global__ __launch_bounds__(256) void deg_k(const int* __restrict__ dst, int* __restrict__ deg) {
    int e = blockIdx.x * 256 + threadIdx.x;
    if (e < NE) atomicAdd(&deg[dst[e]], 1);
}

__global__ __launch_bounds__(256) void norm_k(float* __restrict__ nrm) {
    int i = blockIdx.x * 256 + threadIdx.x;
    if (i < NN) {
        float d = (float)((const int*)nrm)[i];
        nrm[i] = rsqrtf(d < 1.0f ? 1.0f : d);
    }
}

// 16 threads per edge, one float4 (4 feats) per thread: coalesced gather,
// fp32 global atomics scatter (global_atomic_add_f32, no return).
__global__ __launch_bounds__(256) void hop_k(const int* __restrict__ src,
                                             const int* __restrict__ dst,
                                             const float* __restrict__ nrm,
                                             const float* __restrict__ hin,
                                             float* __restrict__ hout) {
    long t = (long)blockIdx.x * 256 + threadIdx.x;
    int  e = (int)(t >> 4);
    if (e >= NE) return;
    int   p  = (int)(t & 15);
    int   s  = src[e];
    int   d  = dst[e];
    float ns = nrm[s];
    float4 v = ((const float4*)(hin + (long)s * F))[p];
    float* op = hout + (long)d * F + p * 4;
    unsafeAtomicAdd(op + 0, v.x * ns);
    unsafeAtomicAdd(op + 1, v.y * ns);
    unsafeAtomicAdd(op + 2, v.z * ns);
    unsafeAtomicAdd(op + 3, v.w * ns);
}

__global__ __launch_bounds__(256) void scale_k(const float* __restrict__ nrm,
                                               float* __restrict__ h) {
    long i = (long)blockIdx.x * 256 + threadIdx.x;
    if (i < (long)NN * F) h[i] *= nrm[i >> 6];
}

// ---------------- WMMA fp32 projection: out = concat @ W.T + b ----------------

// One K=64 segment of the concatenated features.
// A (16x4 f32) per ISA layout: lanes 0-15 carry K=k0,k0+1 ; lanes 16-31 carry K=k0+2,k0+3.
// B (4x16 f32) symmetric: VGPR0 = rows K=k0 (lanes 0-15) / K=k0+2 (lanes 16-31), VGPR1 = +1.
__device__ __forceinline__ void seg_mac(const float* __restrict__ base, int kbase,
                                        int node, int hi2, int ln15,
                                        const float* Wt,
                                        v8f& c0, v8f& c1, v8f& c2, v8f& c3) {
    const float* rp = base + (long)node * F + hi2;
#pragma unroll 4
    for (int kk = 0; kk < F; kk += 4) {
        v2f a;
        a.x = rp[kk];
        a.y = rp[kk + 1];
        int kr = (kbase + kk + hi2) * F + ln15;   // Wt[k][o], o = j*16 + ln15
        v2f b0; b0.x = Wt[kr];      b0.y = Wt[kr + F];
        v2f b1; b1.x = Wt[kr + 16]; b1.y = Wt[kr + F + 16];
        v2f b2; b2.x = Wt[kr + 32]; b2.y = Wt[kr + F + 32];
        v2f b3; b3.x = Wt[kr + 48]; b3.y = Wt[kr + F + 48];
        c0 = __builtin_amdgcn_wmma_f32_16x16x4_f32(false, a, false, b0, (short)0, c0, false, false);
        c1 = __builtin_amdgcn_wmma_f32_16x16x4_f32(false, a, false, b1, (short)0, c1, false, false);
        c2 = __builtin_amdgcn_wmma_f32_16x16x4_f32(false, a, false, b2, (short)0, c2, false, false);
        c3 = __builtin_amdgcn_wmma_f32_16x16x4_f32(false, a, false, b3, (short)0, c3, false, false);
    }
}

__global__ __launch_bounds__(256) void gemm_k(const float* __restrict__ feat,
                                              const float* __restrict__ h1,
                                              const float* __restrict__ h2,
                                              const float* __restrict__ h3,
                                              const float* __restrict__ W,    // [64][256] row-major
                                              const float* __restrict__ bias, // [64]
                                              float* __restrict__ out) {     // [NN][64]
    __shared__ float Wt[KD * F];  // Wt[k][o] = W[o][k], 64 KB
    int tid = threadIdx.x;
    for (int i = tid; i < KD * F; i += 256) {
        int o = i >> 8;     // row of W
        int k = i & 255;    // col of W
        Wt[k * F + o] = W[i];
    }
    __syncthreads();

    int lane = tid & 31;
    int ln15 = lane & 15;
    int hi2  = (lane >= 16) ? 2 : 0;
    int rowBase = blockIdx.x * 128 + (tid >> 5) * 16;
    if (rowBase >= NN) return;           // wave-uniform: EXEC stays all-1s for WMMA

    int node = rowBase + ln15;
    if (node >= NN) node = NN - 1;       // safety clamp (duplicate row, stores guarded)

    v8f c0, c1, c2, c3;
    float bv0 = bias[ln15], bv1 = bias[16 + ln15], bv2 = bias[32 + ln15], bv3 = bias[48 + ln15];
#pragma unroll
    for (int i = 0; i < 8; ++i) { c0[i] = bv0; c1[i] = bv1; c2[i] = bv2; c3[i] = bv3; }

    seg_mac(feat, 0,   node, hi2, ln15, Wt, c0, c1, c2, c3);
    seg_mac(h1,   64,  node, hi2, ln15, Wt, c0, c1, c2, c3);
    seg_mac(h2,   128, node, hi2, ln15, Wt, c0, c1, c2, c3);
    seg_mac(h3,   192, node, hi2, ln15, Wt, c0, c1, c2, c3);

    // C/D layout: VGPR r -> M = r (lanes 0-15) or M = 8+r (lanes 16-31), N = lane&15
    int mBase = rowBase + (hi2 ? 8 : 0);
#pragma unroll
    for (int r = 0; r < 8; ++r) {
        int row = mBase + r;
        if (row < NN) {
            out[(long)row * F + ln15]      = c0[r];
            out[(long)row * F + 16 + ln15] = c1[r];
            out[(long)row * F + 32 + ln15] = c2[r];
            out[(long)row * F + 48 + ln15] = c3[r];
        }
    }
}

// ---------------- launch ----------------

extern "C" void kernel_launch(void* const* d_in, const int* in_sizes, int n_in,
                              void* d_out, int out_size, void* d_ws, size_t ws_size,
                              hipStream_t stream) {
    const float* feat = (const float*)d_in[0];
    const int*   src  = (const int*)d_in[1];
    const int*   dst  = (const int*)d_in[2];
    const float* W    = (const float*)d_in[3];
    const float* bias = (const float*)d_in[4];
    float*       out  = (float*)d_out;

    float* ws   = (float*)d_ws;
    float* nrm  = ws;                      // NN floats (also int deg during counting)
    float* h1   = ws + 50048;              // 64-aligned past NN
    float* h2   = h1 + (long)NN * F;
    float* h3   = h2 + (long)NN * F;
    long   ztot = 50048 + 3L * NN * F;     // ~9.65M floats (~38.6 MB)

    zero_k <<<2048, 256, 0, stream>>>(ws, ztot);
    deg_k  <<<(NE + 255) / 256, 256, 0, stream>>>(dst, (int*)nrm);
    norm_k <<<(NN + 255) / 256, 256, 0, stream>>>(nrm);

    const int hopBlocks   = (NE * 16) / 256;          // 50000
    const int scaleBlocks = (NN * F + 255) / 256;     // 12500

    hop_k  <<<hopBlocks,   256, 0, stream>>>(src, dst, nrm, feat, h1);
    scale_k<<<scaleBlocks, 256, 0, stream>>>(nrm, h1);
    hop_k  <<<hopBlocks,   256, 0, stream>>>(src, dst, nrm, h1, h2);
    scale_k<<<scaleBlocks, 256, 0, stream>>>(nrm, h2);
    hop_k  <<<hopBlocks,   256, 0, stream>>>(src, dst, nrm, h2, h3);
    scale_k<<<scaleBlocks, 256, 0, stream>>>(nrm, h3);

    gemm_k <<<(NN + 127) / 128, 256, 0, stream>>>(feat, h1, h2, h3, W, bias, out);
}